// OptimizedRouter_5033701671230
// MI455X (gfx1250) — compile-verified
//
#include <hip/hip_runtime.h>
#include <hip/hip_bf16.h>

// Problem constants (B=4, S=4096, D=2048, E=8, K=2)
#define DT        2048
#define ET        8
#define NTOK      16384               // B*S
#define NASSIGN   (NTOK * 2)          // 32768 assignments
#define CAPACITY  5120                // int(16384*2/8*1.25)
#define MAXPER    NTOK                // per-expert worst case (top-2 indices distinct)

// d_out layout (floats): [0,32768) indices, [32768,65536) weights, [65536] loss, [65537,98305) mask
#define OFF_W  NASSIGN
#define OFF_L  (2 * NASSIGN)
#define OFF_M  (2 * NASSIGN + 1)

// workspace layout (bytes)
#define WS_CNT 0
#define WS_IMP 64
#define WS_AW  1024
#define WS_AI  (1024 + ET * MAXPER * 4)

typedef float v2f __attribute__((ext_vector_type(2)));
typedef float v8f __attribute__((ext_vector_type(8)));

__device__ __forceinline__ v2f load2(const float* __restrict__ p) {
  float2 t = *reinterpret_cast<const float2*>(p);
  v2f r = {t.x, t.y};
  return r;
}

// Branchless fragment step: B lanes for padded experts read an aliased valid
// row (same cachelines) and are zeroed via v_cndmask — no EXEC branches.
__device__ __forceinline__ void mma_step(const float* __restrict__ a,
                                         const float* __restrict__ b,
                                         float bmask, int k, v8f& acc) {
  v2f av = load2(a + k);
  v2f bt = load2(b + k);
  v2f bv = {bt.x * bmask, bt.y * bmask};
  acc = __builtin_amdgcn_wmma_f32_16x16x4_f32(false, av, false, bv,
                                              (short)0, acc, false, false);
}

__global__ __launch_bounds__(256) void moe_router_kernel(
    const float* __restrict__ x, const float* __restrict__ wg,
    float* __restrict__ out, unsigned* __restrict__ cnt,
    float* __restrict__ imp, float* __restrict__ aw, unsigned* __restrict__ ai) {
  __shared__ float slog[8][16][ET];   // per-wave 16 tokens x 8 expert logits
  __shared__ float simp[ET];

  const int tid  = threadIdx.x;
  const int w    = tid >> 5;
  const int lane = tid & 31;
  const int row  = lane & 15;         // A: token row / B: expert row
  const int koff = (lane >> 4) << 1;  // K columns {0,1} or {2,3}
  const int tokbase = blockIdx.x * 128 + w * 16;

  if (tid < ET) simp[tid] = 0.0f;

  const float* arow  = x  + (size_t)(tokbase + row) * DT + koff;
  const float* brow  = wg + (size_t)(row & 7) * DT + koff;  // alias padded lanes
  const float  bmask = (row < ET) ? 1.0f : 0.0f;

  v8f acc0 = {}; v8f acc1 = {}; v8f acc2 = {}; v8f acc3 = {};
  #pragma unroll 4
  for (int k = 0; k < DT; k += 16) {  // 4 independent WMMA chains for ILP
    mma_step(arow, brow, bmask, k +  0, acc0);
    mma_step(arow, brow, bmask, k +  4, acc1);
    mma_step(arow, brow, bmask, k +  8, acc2);
    mma_step(arow, brow, bmask, k + 12, acc3);
  }
  v8f acc = (acc0 + acc1) + (acc2 + acc3);

  // Stage logits: VGPR r holds M = r (lanes 0-15) / r+8 (lanes 16-31), N = lane&15
  const int mbase = (lane >> 4) << 3;
  if (row < ET) {
    #pragma unroll
    for (int r = 0; r < 8; ++r) slog[w][mbase + r][row] = acc[r];
  }
  __syncthreads();

  if (tid < 128) {                    // one thread per token of this block
    const int g = tid >> 4, m = tid & 15;
    const int tok = blockIdx.x * 128 + tid;

    float p[ET];
    float mx = -3.402823e38f;
    #pragma unroll
    for (int e = 0; e < ET; ++e) { p[e] = slog[g][m][e]; mx = fmaxf(mx, p[e]); }
    float s = 0.0f;
    #pragma unroll
    for (int e = 0; e < ET; ++e) { p[e] = expf(p[e] - mx); s += p[e]; }
    const float inv = 1.0f / s;
    #pragma unroll
    for (int e = 0; e < ET; ++e) p[e] *= inv;

    // top-2 (strict > : first occurrence wins ties, matching lax.top_k)
    float p0 = -1.0f; int i0 = 0;
    #pragma unroll
    for (int e = 0; e < ET; ++e) if (p[e] > p0) { p0 = p[e]; i0 = e; }
    float p1 = -1.0f; int i1 = 0;
    #pragma unroll
    for (int e = 0; e < ET; ++e) if (e != i0 && p[e] > p1) { p1 = p[e]; i1 = e; }
    const float rn = 1.0f / (p0 + p1);
    const float w0 = p0 * rn, w1 = p1 * rn;

    out[tok * 2 + 0] = (float)i0;
    out[tok * 2 + 1] = (float)i1;
    out[OFF_W + tok * 2 + 0] = w0;
    out[OFF_W + tok * 2 + 1] = w1;

    #pragma unroll
    for (int e = 0; e < ET; ++e) atomicAdd(&simp[e], p[e]);

    // scatter assignments into per-expert compacted lists
    unsigned q0 = atomicAdd(&cnt[i0], 1u);
    aw[(size_t)i0 * MAXPER + q0] = w0;
    ai[(size_t)i0 * MAXPER + q0] = (unsigned)(tok * 2 + 0);
    unsigned q1 = atomicAdd(&cnt[i1], 1u);
    aw[(size_t)i1 * MAXPER + q1] = w1;
    ai[(size_t)i1 * MAXPER + q1] = (unsigned)(tok * 2 + 1);
  }
  __syncthreads();
  if (tid < ET) atomicAdd(&imp[tid], simp[tid]);
}

__global__ __launch_bounds__(256) void moe_mask_kernel(
    const float* __restrict__ aw, const unsigned* __restrict__ ai,
    const unsigned* __restrict__ cnt, float* __restrict__ out) {
  const int e = blockIdx.x;
  const int n = (int)cnt[e];
  const float*    wv  = aw + (size_t)e * MAXPER;
  const unsigned* idx = ai + (size_t)e * MAXPER;
  const int tid = threadIdx.x;

  if (n <= CAPACITY) {                // fast path: nothing dropped (uniform branch)
    for (int i = tid; i < n; i += 256) out[OFF_M + idx[i]] = 1.0f;
    return;
  }

  __shared__ float    sw[1024];
  __shared__ unsigned si[1024];
  const int passes = (n + 255) / 256;
  for (int p = 0; p < passes; ++p) {
    const int i = p * 256 + tid;
    const bool act = (i < n);
    float wi = 0.0f; unsigned aii = 0u;
    if (act) { wi = wv[i]; aii = idx[i]; }
    int rank = 0;
    for (int t0 = 0; t0 < n; t0 += 1024) {
      const int tl = min(1024, n - t0);
      __syncthreads();
      for (int j = tid; j < tl; j += 256) { sw[j] = wv[t0 + j]; si[j] = idx[t0 + j]; }
      __syncthreads();
      if (act) {
        for (int j = 0; j < tl; ++j) {
          const float wj = sw[j];
          rank += (wj > wi || (wj == wi && si[j] < aii)) ? 1 : 0;
        }
      }
    }
    if (act) out[OFF_M + aii] = (rank < CAPACITY) ? 1.0f : 0.0f;
  }
}

__global__ void moe_loss_kernel(const unsigned* __restrict__ cnt,
                                const float* __restrict__ imp,
                                float* __restrict__ out) {
  if (threadIdx.x == 0 && blockIdx.x == 0) {
    float isum = 0.0f, csum = 0.0f, iv[ET], cv[ET];
    for (int e = 0; e < ET; ++e) {
      iv[e] = imp[e]; cv[e] = (float)cnt[e];
      isum += iv[e];  csum += cv[e];
    }
    float loss = 0.0f;
    for (int e = 0; e < ET; ++e) loss += (iv[e] / isum) * (cv[e] / csum);
    out[OFF_L] = (float)ET * loss;
  }
}

__global__ void moe_init_kernel(unsigned* __restrict__ cnt, float* __restrict__ imp) {
  if (threadIdx.x < ET) { cnt[threadIdx.x] = 0u; imp[threadIdx.x] = 0.0f; }
}

extern "C" void kernel_launch(void* const* d_in, const int* in_sizes, int n_in,
                              void* d_out, int out_size, void* d_ws, size_t ws_size,
                              hipStream_t stream) {
  const float* x  = (const float*)d_in[0];   // [4,4096,2048] f32
  const float* wg = (const float*)d_in[1];   // [8,2048] f32
  float* out = (float*)d_out;
  char*  ws  = (char*)d_ws;
  unsigned* cnt = (unsigned*)(ws + WS_CNT);
  float*    imp = (float*)(ws + WS_IMP);
  float*    aw  = (float*)(ws + WS_AW);
  unsigned* ai  = (unsigned*)(ws + WS_AI);

  moe_init_kernel<<<1, 32, 0, stream>>>(cnt, imp);
  moe_router_kernel<<<NTOK / 128, 256, 0, stream>>>(x, wg, out, cnt, imp, aw, ai);
  moe_mask_kernel<<<ET, 256, 0, stream>>>(aw, ai, cnt, out);
  moe_loss_kernel<<<1, 32, 0, stream>>>(cnt, imp, out);
}